// YOLOLoss_74071005987436
// MI455X (gfx1250) — compile-verified
//
#include <hip/hip_runtime.h>

// ---------------- types / constants ----------------
typedef __attribute__((ext_vector_type(2))) float v2f;
typedef __attribute__((ext_vector_type(8))) float v8f;

#define NC    80
#define IMGF  640.0f
#define BATCH 16
#define NTGT  32

#define CELLS0 307200   // 16*3*80*80
#define CELLS1 76800    // 16*3*40*40
#define CELLS2 19200    // 16*3*20*20
#define NCELLS 403200
#define BASE1  307200
#define BASE2  384000
#define MAXPOS 2560     // <= B*N*5 dedup'd positive cells

// workspace layout (bytes)
#define OFF_ACC  0                         // 4 floats: obj, cls, box, (pad)
#define OFF_CNT  16                        // 1 u32: positive-cell count
#define OFF_FLAG 256                       // u32[NCELLS] pos flags
#define OFF_LIST (OFF_FLAG + NCELLS * 4)   // u32[MAXPOS] positive cell ids
#define OFF_REC  (OFF_LIST + MAXPOS * 4)   // uint4[NCELLS]: mask0,mask1,mask2,tid
#define WS_BYTES (OFF_REC + NCELLS * 16)   // ~8.08 MB

__device__ __constant__ float d_anch[9][2] = {
  {0.0156f,0.0203f},{0.0250f,0.0469f},{0.0516f,0.0359f},
  {0.0469f,0.0953f},{0.0969f,0.0703f},{0.0922f,0.1859f},
  {0.1813f,0.1406f},{0.2438f,0.3094f},{0.5828f,0.5094f}};

__device__ __forceinline__ float sigm_(float x){ return 1.0f/(1.0f + __expf(-x)); }
__device__ __forceinline__ float bce_(float x, float t){
  return fmaxf(x, 0.0f) - x*t + log1pf(__expf(-fabsf(x)));
}

// 32-lane sum via V_WMMA_F32_16X16X4_F32 (B = ones -> D[m][n] = v_m + v_{m+16}),
// then sum the 8 D VGPRs and fold halves with one swizzle. Requires EXEC all-ones.
__device__ __forceinline__ float wave_sum(float v){
#if defined(__gfx1250__) && __has_builtin(__builtin_amdgcn_wmma_f32_16x16x4_f32)
  v2f a; a[0] = v;    a[1] = 0.0f;
  v2f b; b[0] = 1.0f; b[1] = 1.0f;
  v8f c = {};
  c = __builtin_amdgcn_wmma_f32_16x16x4_f32(false, a, false, b, (short)0, c, false, false);
  float s = c[0]+c[1]+c[2]+c[3]+c[4]+c[5]+c[6]+c[7];
  s += __shfl_xor(s, 16, 32);
  return s;
#else
  for (int off = 16; off > 0; off >>= 1) v += __shfl_xor(v, off, 32);
  return v;
#endif
}

// ---------------- kernel 1: zero workspace ----------------
__global__ void zero_ws_kernel(uint4* __restrict__ p, int n16){
  int i = blockIdx.x * blockDim.x + threadIdx.x;
  if (i < n16){ uint4 z = {0u,0u,0u,0u}; p[i] = z; }
}

// ---------------- kernel 2: per-target assignment + scatter ----------------
__global__ void assign_kernel(const float* __restrict__ tb, const int* __restrict__ tl,
                              unsigned* __restrict__ flags, unsigned* __restrict__ list,
                              unsigned* __restrict__ cnt, unsigned* __restrict__ rec){
  int t = threadIdx.x;                     // 0..511, one thread per target
  if (t >= BATCH * NTGT) return;
  int lab = tl[t];
  if (lab < 0 || lab >= NC) return;        // valid mask

  float x1 = tb[t*4+0], y1 = tb[t*4+1], x2 = tb[t*4+2], y2 = tb[t*4+3];
  float bw = fminf(fmaxf((x2-x1)/IMGF, 1e-6f), 1.0f);
  float bh = fminf(fmaxf((y2-y1)/IMGF, 1e-6f), 1.0f);

  int best = 0; float bestIoU = -1.0f;
  #pragma unroll
  for (int k = 0; k < 9; k++){
    float aw = d_anch[k][0], ah = d_anch[k][1];
    float inter = fminf(bw, aw) * fminf(bh, ah);
    float iou = inter / (bw*bh + aw*ah - inter + 1e-9f);
    if (iou > bestIoU){ bestIoU = iou; best = k; }
  }
  int s = best / 3, a = best % 3;

  float cx = fminf(fmaxf((x1+x2)*0.5f/IMGF, 0.0f), 1.0f - 1e-6f);
  float cy = fminf(fmaxf((y1+y2)*0.5f/IMGF, 0.0f), 1.0f - 1e-6f);

  int w = 80 >> s, h = 80 >> s;
  float gx = cx * (float)w, gy = cy * (float)h;
  int gi = min(max((int)floorf(gx), 0), w - 1);
  int gj = min(max((int)floorf(gy), 0), h - 1);
  float fx = gx - (float)gi, fy = gy - (float)gj;

  const int base = (s == 0) ? 0 : (s == 1 ? BASE1 : BASE2);
  const int dio[5] = {0,-1, 1, 0, 0};
  const int djo[5] = {0, 0, 0,-1, 1};
  bool keep[5];
  keep[0] = true; keep[1] = fx < 0.5f; keep[2] = fx >= 0.5f;
  keep[3] = fy < 0.5f; keep[4] = fy >= 0.5f;

  #pragma unroll
  for (int c = 0; c < 5; c++){
    int ngi = gi + dio[c], ngj = gj + djo[c];
    if (!keep[c] || ngi < 0 || ngi >= w || ngj < 0 || ngj >= h) continue;
    int cell = (((t >> 5) * 3 + a) * h + ngj) * w + ngi;   // b = t>>5
    int g = base + cell;
    unsigned old = atomicOr(&flags[g], 1u);
    if (old == 0u){
      unsigned idx = atomicAdd(cnt, 1u);
      if (idx < MAXPOS) list[idx] = (unsigned)g;
    }
    atomicOr(&rec[(size_t)g*4 + (lab >> 5)], 1u << (lab & 31));
    rec[(size_t)g*4 + 3] = (unsigned)t;    // last-writer-wins target id (matches scatter .set)
  }
}

// ---------------- kernel 3: dense objectness focal-BCE (one channel) ----------------
__global__ void obj_kernel(const float* __restrict__ pred, const unsigned* __restrict__ flags,
                           float* __restrict__ acc, int h, int w, float invN){
  int idx = blockIdx.x * blockDim.x + threadIdx.x;   // exact grid: 48*h*w threads
  int i = idx % w; int r = idx / w;
  int j = r % h;   r /= h;
  int a = r % 3;   int bb = r / 3;
  size_t addr = (((size_t)(bb*255 + a*85 + 4)) * (size_t)h + (size_t)j) * (size_t)w + (size_t)i;
  float x = pred[addr];
  float t = flags[idx] ? 1.0f : 0.0f;
  float p  = sigm_(x);
  float bl = bce_(x, t);
  float pt = p*t + (1.0f - p)*(1.0f - t);
  float at = 0.25f*t + 0.75f*(1.0f - t);
  float om = 1.0f - pt;
  float partial = at * om * om * bl * invN;           // LAM_OBJ == LAM_NOOBJ == 1
  float s = wave_sum(partial);
  if ((threadIdx.x & 31) == 0) atomicAdd(acc + 0, s);
}

// ---------------- kernel 4: sparse positive cells (box GIoU+L1, cls BCE) ----------------
__global__ void pos_kernel(const float* __restrict__ p0, const float* __restrict__ p1,
                           const float* __restrict__ p2, const float* __restrict__ tb,
                           const unsigned* __restrict__ list, const unsigned* __restrict__ cnt,
                           const uint4* __restrict__ rec, float* __restrict__ acc){
  int idx = blockIdx.x * blockDim.x + threadIdx.x;
  unsigned n = *cnt; if (n > MAXPOS) n = MAXPOS;
  float boxp = 0.0f, clsp = 0.0f;
  if (idx < (int)n){
    unsigned g = list[idx];
    int s = (g >= BASE2) ? 2 : ((g >= BASE1) ? 1 : 0);
    const float* pred = (s == 0) ? p0 : ((s == 1) ? p1 : p2);
    int w = 80 >> s, h = 80 >> s;
    int base = (s == 0) ? 0 : (s == 1 ? BASE1 : BASE2);
    int cell = (int)g - base;
    int i = cell % w; int r = cell / w;
    int j = r % h;    r /= h;
    int a = r % 3;    int bb = r / 3;

    uint4 rc = rec[g];
    int t = (int)rc.w;
    // recompute target box from target id (keeps record at 16B)
    float x1 = tb[t*4+0], y1 = tb[t*4+1], x2 = tb[t*4+2], y2 = tb[t*4+3];
    float tw = fminf(fmaxf((x2-x1)/IMGF, 1e-6f), 1.0f);
    float th = fminf(fmaxf((y2-y1)/IMGF, 1e-6f), 1.0f);
    float tcx = fminf(fmaxf((x1+x2)*0.5f/IMGF, 0.0f), 1.0f - 1e-6f);
    float tcy = fminf(fmaxf((y1+y2)*0.5f/IMGF, 0.0f), 1.0f - 1e-6f);

    size_t hw = (size_t)h * (size_t)w;
    size_t basea = ((size_t)(bb*255 + a*85)) * hw + (size_t)j * (size_t)w + (size_t)i;

    float sx = sigm_(pred[basea + 0*hw]);
    float sy = sigm_(pred[basea + 1*hw]);
    float sw = sigm_(pred[basea + 2*hw]);
    float sh = sigm_(pred[basea + 3*hw]);
    float pcx = ((float)i + 2.0f*sx - 0.5f) / (float)w;
    float pcy = ((float)j + 2.0f*sy - 0.5f) / (float)h;
    float pw = (2.0f*sw)*(2.0f*sw) * d_anch[s*3 + a][0];
    float ph = (2.0f*sh)*(2.0f*sh) * d_anch[s*3 + a][1];

    float px1 = pcx - pw*0.5f, px2 = pcx + pw*0.5f;
    float py1 = pcy - ph*0.5f, py2 = pcy + ph*0.5f;
    float tx1 = tcx - tw*0.5f, tx2 = tcx + tw*0.5f;
    float ty1 = tcy - th*0.5f, ty2 = tcy + th*0.5f;
    float iw = fmaxf(fminf(px2, tx2) - fmaxf(px1, tx1), 0.0f);
    float ih = fmaxf(fminf(py2, ty2) - fmaxf(py1, ty1), 0.0f);
    float it = iw * ih;
    float un = pw*ph + tw*th - it + 1e-9f;
    float iou = it / un;
    float enc = (fmaxf(px2, tx2) - fminf(px1, tx1)) * (fmaxf(py2, ty2) - fminf(py1, ty1)) + 1e-9f;
    float giou = iou - (enc - un) / enc;
    float l1 = fabsf(pcx-tcx) + fabsf(pcy-tcy) + fabsf(pw-tw) + fabsf(ph-th);
    boxp = 1.0f - giou + l1;

    float cs = 0.0f;
    #pragma unroll 4
    for (int c = 0; c < NC; c++){
      float x = pred[basea + (size_t)(5 + c) * hw];
      unsigned mw = (c < 32) ? rc.x : ((c < 64) ? rc.y : rc.z);
      float tt = (float)((mw >> (c & 31)) & 1u);
      cs += bce_(x, tt);
    }
    clsp = cs;
  }
  // all lanes (EXEC full) participate in the WMMA reductions
  float bs = wave_sum(boxp);
  float cs2 = wave_sum(clsp);
  if ((threadIdx.x & 31) == 0){
    atomicAdd(acc + 2, bs);
    atomicAdd(acc + 1, cs2);
  }
}

// ---------------- kernel 5: combine ----------------
__global__ void final_kernel(const float* __restrict__ acc, const unsigned* __restrict__ cnt,
                             float* __restrict__ out){
  if (threadIdx.x == 0 && blockIdx.x == 0){
    float npos = fmaxf((float)(*cnt), 1.0f);
    out[0] = acc[0] + (1.0f * acc[1] + 5.0f * acc[2]) / npos;
  }
}

// ---------------- launcher ----------------
extern "C" void kernel_launch(void* const* d_in, const int* in_sizes, int n_in,
                              void* d_out, int out_size, void* d_ws, size_t ws_size,
                              hipStream_t stream){
  const float* p0 = (const float*)d_in[0];
  const float* p1 = (const float*)d_in[1];
  const float* p2 = (const float*)d_in[2];
  const float* tb = (const float*)d_in[3];
  const int*   tl = (const int*)d_in[4];
  char* ws = (char*)d_ws;

  float*    acc   = (float*)(ws + OFF_ACC);
  unsigned* cnt   = (unsigned*)(ws + OFF_CNT);
  unsigned* flags = (unsigned*)(ws + OFF_FLAG);
  unsigned* list  = (unsigned*)(ws + OFF_LIST);
  unsigned* recw  = (unsigned*)(ws + OFF_REC);
  uint4*    rec4  = (uint4*)(ws + OFF_REC);

  size_t zb = (ws_size < (size_t)WS_BYTES) ? ws_size : (size_t)WS_BYTES;
  int n16 = (int)(zb / 16);
  zero_ws_kernel<<<(n16 + 255)/256, 256, 0, stream>>>((uint4*)ws, n16);

  assign_kernel<<<1, BATCH*NTGT, 0, stream>>>(tb, tl, flags, list, cnt, recw);

  // dense objectness pass per scale (exact grids: cell counts are multiples of 256)
  obj_kernel<<<CELLS0/256, 256, 0, stream>>>(p0, flags,         acc, 80, 80, 1.0f/(float)CELLS0);
  obj_kernel<<<CELLS1/256, 256, 0, stream>>>(p1, flags + BASE1, acc, 40, 40, 1.0f/(float)CELLS1);
  obj_kernel<<<CELLS2/256, 256, 0, stream>>>(p2, flags + BASE2, acc, 20, 20, 1.0f/(float)CELLS2);

  pos_kernel<<<MAXPOS/256, 256, 0, stream>>>(p0, p1, p2, tb, list, cnt, rec4, acc);

  final_kernel<<<1, 1, 0, stream>>>(acc, cnt, (float*)d_out);
}